// Explainer_90744069030452
// MI455X (gfx1250) — compile-verified
//
#include <hip/hip_runtime.h>
#include <math.h>

#define NN   1024
#define DEMB 64
#define FIN  128
#define NCLS 8
#define HH   64

typedef float v2f __attribute__((ext_vector_type(2)));
typedef float v8f __attribute__((ext_vector_type(8)));

__device__ __forceinline__ v8f wmma4(v2f a, v2f b, v8f c) {
  // V_WMMA_F32_16X16X4_F32: D = A(16x4) * B(4x16) + C, fp32 exact
  return __builtin_amdgcn_wmma_f32_16x16x4_f32(
      /*neg_a=*/false, a, /*neg_b=*/false, b,
      /*c_mod=*/(short)0, c, /*reuse_a=*/false, /*reuse_b=*/false);
}

// One wave (32 lanes) computes one 16x16 fp32 tile: C = A(16xK) * B(Kx16).
// ISA layout: A lane m (m=lane&15) holds K = k0 + 2*(lane>>4) + {0,1};
//             B lane holds row K = k0 + 2*(lane>>4) + {0,1}, col = lane&15.
__device__ __forceinline__ v8f wave_gemm16x16(const float* __restrict__ A, int lda,
                                              const float* __restrict__ B, int ldb,
                                              int K) {
  const int lane = threadIdx.x & 31;
  const int half = lane >> 4;
  const int lm   = lane & 15;
  v8f c = {};
  const float* ap = A + (size_t)lm * lda + 2 * half;
#pragma unroll 4
  for (int k0 = 0; k0 < K; k0 += 4) {
    v2f a;
    a.x = ap[k0 + 0];
    a.y = ap[k0 + 1];
    const int kk = k0 + 2 * half;
    v2f b;
    b.x = B[(size_t)kk * ldb + lm];
    b.y = B[(size_t)(kk + 1) * ldb + lm];
    c = wmma4(a, b, c);
  }
  return c;
}

// C/D layout: VGPR i -> row (lane<16 ? i : 8+i), col = lane&15
__device__ __forceinline__ void store_tile(float* __restrict__ C, int ldc, v8f c) {
  const int lane = threadIdx.x & 31;
  const int half = lane >> 4;
  const int lm   = lane & 15;
#pragma unroll
  for (int i = 0; i < 8; ++i)
    C[(size_t)(half * 8 + i) * ldc + lm] = c[i];
}

__global__ void k_zero(float* __restrict__ p, int n) {
  int i = blockIdx.x * blockDim.x + threadIdx.x;
  if (i < n) p[i] = 0.0f;
}

// AB[n, 0:64]  = embed @ W1[0:64,:]   (A-part)
// AB[n,64:128] = embed @ W1[64:128,:] (B-part)
__global__ void k_embed_mlp(const float* __restrict__ embed,
                            const float* __restrict__ W1,
                            float* __restrict__ AB) {
  const int nb = blockIdx.x * 16;  // 0..112 (8 tiles over 128 cols)
  const int mb = blockIdx.y * 16;  // 0..1008
  const float* Bm = (nb < HH) ? (W1 + nb) : (W1 + HH * HH + (nb - HH));
  v8f c = wave_gemm16x16(embed + (size_t)mb * DEMB, DEMB, Bm, HH, DEMB);
  store_tile(AB + (size_t)mb * (2 * HH) + nb, 2 * HH, c);
}

// gate[r,c] = sigmoid((log(u) - log1p(-u) + logalpha(r,c)) / beta)
// logalpha(r,c) = sum_k relu(A[r,k]+B[c,k]+b1[k]) * W2[k] + b2
__global__ void k_gate(const float* __restrict__ AB,
                       const float* __restrict__ b1,
                       const float* __restrict__ W2,
                       const float* __restrict__ b2,
                       const float* __restrict__ tmp,
                       const float* __restrict__ noise,
                       float* __restrict__ gate) {
  __shared__ float As[16][HH + 1];   // +1 pad: avoid 16-way LDS bank conflicts
  __shared__ float Bs[16][HH + 1];
  __shared__ float b1s[HH];
  __shared__ float w2s[HH];
  const int tx = threadIdx.x, ty = threadIdx.y;
  const int tid = ty * 16 + tx;
  const int rb = blockIdx.y * 16, cb = blockIdx.x * 16;

  for (int idx = tid; idx < 16 * HH; idx += 256) {
    const int rr = idx >> 6, kk = idx & 63;
    As[rr][kk] = AB[(size_t)(rb + rr) * (2 * HH) + kk];
    Bs[rr][kk] = AB[(size_t)(cb + rr) * (2 * HH) + HH + kk];
  }
  if (tid < HH) { b1s[tid] = b1[tid]; w2s[tid] = W2[tid]; }
  __syncthreads();

  float la = b2[0];
#pragma unroll 8
  for (int k = 0; k < HH; ++k) {
    const float h = As[ty][k] + Bs[tx][k] + b1s[k];
    la += fmaxf(h, 0.0f) * w2s[k];
  }
  const float beta = tmp[0];
  const size_t nidx = (size_t)(rb + ty) * NN + (cb + tx);
  const float u = noise[nidx];
  const float z = (__logf(u) - log1pf(-u) + la) / beta;
  gate[nidx] = 1.0f / (1.0f + __expf(-z));
}

// madj = adj * 0.5*(gate + gate^T)
__global__ void k_mask(const float* __restrict__ adj,
                       const float* __restrict__ gate,
                       float* __restrict__ madj) {
  const size_t i = (size_t)blockIdx.x * blockDim.x + threadIdx.x;
  const int r = (int)(i >> 10), c = (int)(i & (NN - 1));
  const float g  = gate[i];
  const float gt = gate[(size_t)c * NN + r];
  madj[i] = adj[i] * 0.5f * (g + gt);
}

// Y = madj @ x  : [1024,1024] x [1024,128]
__global__ void k_aggregate(const float* __restrict__ madj,
                            const float* __restrict__ x,
                            float* __restrict__ Y) {
  const int nb = blockIdx.x * 16;  // over 128 cols
  const int mb = blockIdx.y * 16;
  v8f c = wave_gemm16x16(madj + (size_t)mb * NN, NN, x + nb, FIN, NN);
  store_tile(Y + (size_t)mb * FIN + nb, FIN, c);
}

// colsum += column sums of relu(Y @ Wg1)  (mean pool, Z never materialized)
__global__ void k_pool(const float* __restrict__ Y,
                       const float* __restrict__ Wg1,
                       float* __restrict__ colsum) {
  const int nb = blockIdx.x * 16;  // over 64 cols
  const int mb = blockIdx.y * 16;
  v8f c = wave_gemm16x16(Y + (size_t)mb * FIN, FIN, Wg1 + nb, HH, FIN);
  float acc = 0.0f;
#pragma unroll
  for (int i = 0; i < 8; ++i) acc += fmaxf(c[i], 0.0f);
  atomicAdd(&colsum[nb + (threadIdx.x & 15)], acc);
}

// logits = (colsum/N) @ Wg2 ; softmax -> out[8]
__global__ void k_head(const float* __restrict__ colsum,
                       const float* __restrict__ Wg2,
                       float* __restrict__ out) {
  __shared__ float lg[NCLS];
  const int t = threadIdx.x;
  if (t < NCLS) {
    float l = 0.0f;
    for (int j = 0; j < HH; ++j)
      l += (colsum[j] * (1.0f / NN)) * Wg2[j * NCLS + t];
    lg[t] = l;
  }
  __syncthreads();
  if (t < NCLS) {
    float m = lg[0];
    for (int j = 1; j < NCLS; ++j) m = fmaxf(m, lg[j]);
    float s = 0.0f;
    for (int j = 0; j < NCLS; ++j) s += __expf(lg[j] - m);
    out[t] = __expf(lg[t] - m) / s;
  }
}

extern "C" void kernel_launch(void* const* d_in, const int* in_sizes, int n_in,
                              void* d_out, int out_size, void* d_ws, size_t ws_size,
                              hipStream_t stream) {
  const float* x     = (const float*)d_in[0];   // [1024,128]
  const float* embed = (const float*)d_in[1];   // [1024,64]
  const float* adj   = (const float*)d_in[2];   // [1024,1024]
  const float* tmp   = (const float*)d_in[3];   // [1] beta
  const float* noise = (const float*)d_in[4];   // [1024*1024]
  /* label d_in[5] unused by reference */
  const float* W1    = (const float*)d_in[6];   // [128,64]
  const float* b1    = (const float*)d_in[7];   // [64]
  const float* W2    = (const float*)d_in[8];   // [64]
  const float* b2    = (const float*)d_in[9];   // [1]
  const float* Wg1   = (const float*)d_in[10];  // [128,64]
  const float* Wg2   = (const float*)d_in[11];  // [64,8]

  float* ws     = (float*)d_ws;
  float* AB     = ws;                                  // 1024*128
  float* gate   = AB + (size_t)NN * 2 * HH;            // 1024*1024
  float* madj   = gate + (size_t)NN * NN;              // 1024*1024
  float* Y      = madj + (size_t)NN * NN;              // 1024*128
  float* colsum = Y + (size_t)NN * FIN;                // 64

  k_zero<<<1, 64, 0, stream>>>(colsum, HH);
  k_embed_mlp<<<dim3(2 * HH / 16, NN / 16), 32, 0, stream>>>(embed, W1, AB);
  k_gate<<<dim3(NN / 16, NN / 16), dim3(16, 16), 0, stream>>>(AB, b1, W2, b2, tmp, noise, gate);
  k_mask<<<dim3((NN * NN) / 256), 256, 0, stream>>>(adj, gate, madj);
  k_aggregate<<<dim3(FIN / 16, NN / 16), 32, 0, stream>>>(madj, x, Y);
  k_pool<<<dim3(HH / 16, NN / 16), 32, 0, stream>>>(Y, Wg1, colsum);
  k_head<<<1, 32, 0, stream>>>(colsum, Wg2, (float*)d_out);
}